// LiDARTokenizer_64166811402568
// MI455X (gfx1250) — compile-verified
//
#include <hip/hip_runtime.h>
#include <hip/hip_bf16.h>
#include <math.h>

// ---------------------------------------------------------------------------
// Problem constants (match setup_inputs exactly)
// ---------------------------------------------------------------------------
#define BATCH   32
#define NPTS    131072
#define CIN     6
#define MSAMP   1024
#define DMODEL  256

typedef float v2f __attribute__((ext_vector_type(2)));
typedef float v8f __attribute__((ext_vector_type(8)));

// ---------------------------------------------------------------------------
// Kernel 1: transpose AoS [B,N,6] xyz -> SoA planes x[B*N], y[B*N], z[B*N]
// (coalesced streaming for the FPS hot loop; done once per launch, ~150MB)
// ---------------------------------------------------------------------------
__global__ __launch_bounds__(256) void xyz_transpose_kernel(
    const float* __restrict__ pts, float* __restrict__ sx,
    float* __restrict__ sy, float* __restrict__ sz) {
  size_t i = (size_t)blockIdx.x * blockDim.x + threadIdx.x;
  if (i >= (size_t)BATCH * NPTS) return;
  const float* p = pts + i * CIN;
  sx[i] = p[0];
  sy[i] = p[1];
  sz[i] = p[2];
}

// ---------------------------------------------------------------------------
// Kernel 2: farthest point sampling. One workgroup (1024 thr = 32 waves) per
// batch; dist lives in L2-resident workspace; per-step block argmax via
// wave32 shuffles + LDS cross-wave stage. Winning point's coords travel
// through the reduction so the next step starts without a dependent load.
// ---------------------------------------------------------------------------
__global__ __launch_bounds__(1024) void fps_kernel(
    const float* __restrict__ sx, const float* __restrict__ sy,
    const float* __restrict__ sz, float* __restrict__ dist,
    int* __restrict__ sel) {
  const int b   = blockIdx.x;
  const int tid = threadIdx.x;
  const float* x = sx + (size_t)b * NPTS;
  const float* y = sy + (size_t)b * NPTS;
  const float* z = sz + (size_t)b * NPTS;
  float* db = dist + (size_t)b * NPTS;

  __shared__ float s_val[32], s_px[32], s_py[32], s_pz[32];
  __shared__ int   s_idx[32];
  __shared__ float s_cx, s_cy, s_cz;
  __shared__ int   s_cur;

  int cur = 0;
  if (tid == 0) { s_cx = x[0]; s_cy = y[0]; s_cz = z[0]; }
  __syncthreads();

  const int PER = NPTS / 1024;  // 128 points per thread, stride-1024 (coalesced)

  for (int s = 0; s < MSAMP; ++s) {
    if (tid == 0) sel[b * MSAMP + s] = cur;
    const float cx = s_cx, cy = s_cy, cz = s_cz;

    float best = -1.0f, bx = 0.f, by = 0.f, bz = 0.f;
    int   bi = 0;

    if (s == 0) {
      // dist is uninitialized (poisoned ws): treat old = +inf, never read it.
#pragma unroll 4
      for (int j = 0; j < PER; ++j) {
        int p = j * 1024 + tid;
        float px = x[p], py = y[p], pz = z[p];
        float dx = px - cx, dy = py - cy, dz = pz - cz;
        float d = dx * dx + dy * dy + dz * dz;
        db[p] = d;
        if (d > best) { best = d; bi = p; bx = px; by = py; bz = pz; }
      }
    } else {
#pragma unroll 4
      for (int j = 0; j < PER; ++j) {
        int p = j * 1024 + tid;
        float px = x[p], py = y[p], pz = z[p];
        float dx = px - cx, dy = py - cy, dz = pz - cz;
        float d = dx * dx + dy * dy + dz * dz;
        float o = db[p];
        d = fminf(o, d);
        db[p] = d;
        if (d > best) { best = d; bi = p; bx = px; by = py; bz = pz; }
      }
    }

    // wave32 argmax reduction (prefer lower index on ties -> jnp.argmax semantics)
#pragma unroll
    for (int off = 16; off > 0; off >>= 1) {
      float ov = __shfl_down(best, off, 32);
      int   oi = __shfl_down(bi,   off, 32);
      float ox = __shfl_down(bx,   off, 32);
      float oy = __shfl_down(by,   off, 32);
      float oz = __shfl_down(bz,   off, 32);
      if (ov > best || (ov == best && oi < bi)) {
        best = ov; bi = oi; bx = ox; by = oy; bz = oz;
      }
    }
    const int wv = tid >> 5;
    if ((tid & 31) == 0) {
      s_val[wv] = best; s_idx[wv] = bi;
      s_px[wv] = bx; s_py[wv] = by; s_pz[wv] = bz;
    }
    __syncthreads();

    if (tid < 32) {  // wave 0: reduce the 32 wave winners
      float v = s_val[tid]; int ii = s_idx[tid];
      float px = s_px[tid], py = s_py[tid], pz = s_pz[tid];
#pragma unroll
      for (int off = 16; off > 0; off >>= 1) {
        float ov = __shfl_down(v,  off, 32);
        int   oi = __shfl_down(ii, off, 32);
        float ox = __shfl_down(px, off, 32);
        float oy = __shfl_down(py, off, 32);
        float oz = __shfl_down(pz, off, 32);
        if (ov > v || (ov == v && oi < ii)) {
          v = ov; ii = oi; px = ox; py = oy; pz = oz;
        }
      }
      if (tid == 0) { s_cur = ii; s_cx = px; s_cy = py; s_cz = pz; }
    }
    __syncthreads();
    cur = s_cur;
  }
}

// ---------------------------------------------------------------------------
// Kernel 3: gather + 3-layer MLP via V_WMMA_F32_16X16X4_F32 + PE.
// One wave per 16-row tile. Intermediates round-trip through padded per-wave
// LDS scratch to convert the C/D (row-striped-across-lanes) layout to the
// A (row-per-lane) layout. K-padding is branch-free: the padded K rows carry
// A == 0, so the matching B lanes may hold any finite value (we re-load
// valid W1 rows unconditionally instead of branching per half-wave).
// ---------------------------------------------------------------------------
#define WAVES_PER_BLK 4
#define HA_STRIDE 65   // 16x64 buffer, pad to dodge bank conflicts on column reads
#define HB_STRIDE 129  // 16x128 buffer

__global__ __launch_bounds__(128) void tokenizer_kernel(
    const float* __restrict__ pts, const int* __restrict__ sel,
    const float* __restrict__ W1, const float* __restrict__ b1,
    const float* __restrict__ W2, const float* __restrict__ b2,
    const float* __restrict__ W3, const float* __restrict__ b3,
    float* __restrict__ out) {
  __shared__ float hA[WAVES_PER_BLK][16 * HA_STRIDE];
  __shared__ float hB[WAVES_PER_BLK][16 * HB_STRIDE];

  const int lane = threadIdx.x & 31;
  const int wave = threadIdx.x >> 5;
  const int tile = blockIdx.x * WAVES_PER_BLK + wave;  // 0..2047
  const int r0   = tile * 16;
  const int h    = lane >> 4;  // half-wave select (K-pair / row+8 select)
  const int lr   = lane & 15;

  float* HA = &hA[wave][0];
  float* HB = &hB[wave][0];

  // ---- gather this lane's input row (lanes l and l+16 share a row) ----
  const int row = r0 + lr;           // 0..32767
  const int bb  = row >> 10;         // batch
  const int pi  = sel[row];
  const float* prow = pts + ((size_t)bb * NPTS + pi) * CIN;
  const float p0 = prow[0], p1 = prow[1], p2 = prow[2];
  const float p3 = prow[3], p4 = prow[4], p5 = prow[5];

  // A fragments for 16x16x4 f32 WMMA: lanes 0-15 hold K={k0,k0+1}, 16-31 K={k0+2,k0+3}
  v2f a0, a1;
  a0.x = h ? p2 : p0;  a0.y = h ? p3 : p1;      // k0 = 0
  a1.x = h ? 0.f : p4; a1.y = h ? 0.f : p5;     // k0 = 4 (K=6,7 zero in A)

  // ---- Layer 1: [16x8] x [8x64] -> relu -> HA ----
  for (int nt = 0; nt < 4; ++nt) {
    const int n0 = nt * 16;
    v8f acc = {};
    v2f bf;
    bf.x = W1[(0 + 2 * h) * 64 + n0 + lr];
    bf.y = W1[(1 + 2 * h) * 64 + n0 + lr];
    acc = __builtin_amdgcn_wmma_f32_16x16x4_f32(false, a0, false, bf, (short)0,
                                                acc, false, false);
    // K=6,7 lanes multiply against A==0, so re-using rows 4,5 of W1 there is
    // harmless and keeps the load unconditional (no exec-mask branch).
    bf.x = W1[4 * 64 + n0 + lr];
    bf.y = W1[5 * 64 + n0 + lr];
    acc = __builtin_amdgcn_wmma_f32_16x16x4_f32(false, a1, false, bf, (short)0,
                                                acc, false, false);
    const float bias = b1[n0 + lr];
#pragma unroll
    for (int r = 0; r < 8; ++r) {
      float v = acc[r] + bias;
      v = v > 0.f ? v : 0.f;
      HA[(r + 8 * h) * HA_STRIDE + n0 + lr] = v;
    }
  }

  // ---- Layer 2: [16x64] x [64x128] -> relu -> HB ----
  for (int nt = 0; nt < 8; ++nt) {
    const int n0 = nt * 16;
    v8f acc = {};
#pragma unroll
    for (int k0 = 0; k0 < 64; k0 += 4) {
      v2f af, bf;
      af.x = HA[lr * HA_STRIDE + k0 + 2 * h];
      af.y = HA[lr * HA_STRIDE + k0 + 1 + 2 * h];
      bf.x = W2[(k0 + 2 * h) * 128 + n0 + lr];
      bf.y = W2[(k0 + 1 + 2 * h) * 128 + n0 + lr];
      acc = __builtin_amdgcn_wmma_f32_16x16x4_f32(false, af, false, bf, (short)0,
                                                  acc, false, false);
    }
    const float bias = b2[n0 + lr];
#pragma unroll
    for (int r = 0; r < 8; ++r) {
      float v = acc[r] + bias;
      v = v > 0.f ? v : 0.f;
      HB[(r + 8 * h) * HB_STRIDE + n0 + lr] = v;
    }
  }

  // ---- Layer 3: [16x128] x [128x256] + bias + sinusoidal PE -> out ----
  const float kExp = (float)(-9.210340371976184 / 256.0);  // -ln(10000)/d_model
  for (int nt = 0; nt < 16; ++nt) {
    const int n0  = nt * 16;
    const int col = n0 + lr;
    v8f acc = {};
#pragma unroll 8
    for (int k0 = 0; k0 < 128; k0 += 4) {
      v2f af, bf;
      af.x = HB[lr * HB_STRIDE + k0 + 2 * h];
      af.y = HB[lr * HB_STRIDE + k0 + 1 + 2 * h];
      bf.x = W3[(k0 + 2 * h) * DMODEL + col];
      bf.y = W3[(k0 + 1 + 2 * h) * DMODEL + col];
      acc = __builtin_amdgcn_wmma_f32_16x16x4_f32(false, af, false, bf, (short)0,
                                                  acc, false, false);
    }
    const float bias   = b3[col];
    const float factor = expf((float)(col & ~1) * kExp);  // div[col/2]
#pragma unroll
    for (int r = 0; r < 8; ++r) {
      const int rr  = r0 + r + 8 * h;   // global row
      const int m   = rr & (MSAMP - 1); // position within batch
      const float ang = (float)m * factor;
      const float pe  = (col & 1) ? cosf(ang) : sinf(ang);
      out[(size_t)rr * DMODEL + col] = acc[r] + bias + pe;
    }
  }
}

// ---------------------------------------------------------------------------
// Launcher
// ---------------------------------------------------------------------------
extern "C" void kernel_launch(void* const* d_in, const int* in_sizes, int n_in,
                              void* d_out, int out_size, void* d_ws,
                              size_t ws_size, hipStream_t stream) {
  const float* points = (const float*)d_in[0];
  const float* W1 = (const float*)d_in[1];
  const float* b1 = (const float*)d_in[2];
  const float* W2 = (const float*)d_in[3];
  const float* b2 = (const float*)d_in[4];
  const float* W3 = (const float*)d_in[5];
  const float* b3 = (const float*)d_in[6];
  float* out = (float*)d_out;

  const size_t BN = (size_t)BATCH * NPTS;
  float* sx   = (float*)d_ws;
  float* sy   = sx + BN;
  float* sz   = sy + BN;
  float* dist = sz + BN;
  int*   sel  = (int*)(dist + BN);

  // 1) one-time SoA transpose of xyz (coalesced FPS streaming)
  {
    const int thr = 256;
    const int blk = (int)((BN + thr - 1) / thr);
    xyz_transpose_kernel<<<blk, thr, 0, stream>>>(points, sx, sy, sz);
  }
  // 2) farthest point sampling: one WGP-filling workgroup per batch
  fps_kernel<<<BATCH, 1024, 0, stream>>>(sx, sy, sz, dist, sel);
  // 3) gather + WMMA MLP + positional encoding
  {
    const int tiles  = (BATCH * MSAMP) / 16;           // 2048
    const int blocks = tiles / WAVES_PER_BLK;          // 512
    tokenizer_kernel<<<blocks, 128, 0, stream>>>(points, sel, W1, b1, W2, b2,
                                                 W3, b3, out);
  }
}